// kmeans_67181878444598
// MI455X (gfx1250) — compile-verified
//
#include <hip/hip_runtime.h>

typedef __attribute__((ext_vector_type(2))) float v2f;
typedef __attribute__((ext_vector_type(8))) float v8f;

#define N_TOTAL 131072
#define D_DIM   256
#define K_DIM   16
#define ROWS_PER_BLOCK 1024
#define NB      (N_TOTAL / ROWS_PER_BLOCK)   // 128 blocks per pair
#define WAVES   16
#define THREADS (WAVES * 32)                 // 512 threads, 16 wave32

// Kernel 1: per-block partial FTh (16x256 tile) via V_WMMA_F32_16X16X4_F32,
// plus per-block partial sum(h*h). blockIdx.z selects (h_real,F_mat) vs (h_aug,F_aug).
__global__ __launch_bounds__(THREADS)
void kmeans_fth_partial_kernel(const float* __restrict__ h_real,
                               const float* __restrict__ h_aug,
                               const float* __restrict__ F_mat,
                               const float* __restrict__ F_aug,
                               float* __restrict__ fth_ws,   // [2][NB][16*256]
                               float* __restrict__ qss_ws)   // [2][NB]
{
    const int pair = blockIdx.z;
    const float* __restrict__ h = pair ? h_aug : h_real;
    const float* __restrict__ F = pair ? F_aug : F_mat;

    const int tid  = threadIdx.x;
    const int lane = tid & 31;
    const int wave = tid >> 5;            // 0..15 : owns d-tile [16*wave, 16*wave+16)
    const int m    = lane & 15;           // A: row k ; B: column within tile
    const int jb   = (lane >> 4) << 1;    // 0 or 2 : which pair of K-slices (n rows)
    const int d0   = wave * 16;

    const int n_base = blockIdx.x * ROWS_PER_BLOCK;

    // Per-lane base pointers (row n_base + jb); advance by 4 rows per iteration.
    const float* fp = F + (size_t)(n_base + jb) * K_DIM + m;
    const float* hp = h + (size_t)(n_base + jb) * D_DIM + d0 + m;

    v8f acc = {};
    float ss = 0.0f;

#pragma unroll 4
    for (int t = 0; t < ROWS_PER_BLOCK / 4; ++t) {
        v2f a, b;
        // A fragment: A[m, jb] , A[m, jb+1]  with A[k,j] = F[n0+j, k]
        a.x = fp[0];
        a.y = fp[K_DIM];
        // B fragment: B[jb, m] , B[jb+1, m]  with B[j,d] = h[n0+j, d0+d]
        b.x = hp[0];
        b.y = hp[D_DIM];

        ss += b.x * b.x + b.y * b.y;      // every h element counted exactly once

        // v_wmma_f32_16x16x4_f32 : acc(16x16 f32) += A(16x4 f32) * B(4x16 f32)
        acc = __builtin_amdgcn_wmma_f32_16x16x4_f32(
                  /*neg_a=*/false, a, /*neg_b=*/false, b,
                  /*c_mod=*/(short)0, acc, /*reuse_a=*/false, /*reuse_b=*/false);

        fp += 4 * K_DIM;
        hp += 4 * D_DIM;
    }

    // Write partial FTh tile: D-matrix VGPR r, lanes 0-15 -> M=r, lanes 16-31 -> M=r+8
    {
        float* dst = fth_ws + ((size_t)pair * NB + blockIdx.x) * (K_DIM * D_DIM);
        const int kbase = (lane >> 4) << 3;   // 0 or 8
        const int d     = d0 + m;
#pragma unroll
        for (int r = 0; r < 8; ++r) {
            dst[(size_t)(kbase + r) * D_DIM + d] = acc[r];
        }
    }

    // Block-wide reduction of sum(h*h)
    __shared__ float red[THREADS];
    red[tid] = ss;
    __syncthreads();
#pragma unroll
    for (int s = THREADS / 2; s > 0; s >>= 1) {
        if (tid < s) red[tid] += red[tid + s];
        __syncthreads();
    }
    if (tid == 0) qss_ws[pair * NB + blockIdx.x] = red[0];
}

// Kernel 2: one block per pair. Sum partial FTh tiles over blocks, square+sum,
// sum partial ||h||^2, emit loss = sumsq(h) - sumsq(FTh).
__global__ __launch_bounds__(256)
void kmeans_reduce_kernel(const float* __restrict__ fth_ws,
                          const float* __restrict__ qss_ws,
                          float* __restrict__ out)
{
    const int pair = blockIdx.x;
    const int tid  = threadIdx.x;
    const float* fth = fth_ws + (size_t)pair * NB * (K_DIM * D_DIM);
    const float* qss = qss_ws + pair * NB;

    float S = 0.0f;                        // sum of squares of fully-reduced FTh
    for (int e = tid; e < K_DIM * D_DIM; e += 256) {
        float v = 0.0f;
        for (int b = 0; b < NB; ++b) v += fth[(size_t)b * (K_DIM * D_DIM) + e];
        S += v * v;
    }
    float Q = 0.0f;                        // total sum(h*h)
    for (int b = tid; b < NB; b += 256) Q += qss[b];

    __shared__ float sS[256];
    __shared__ float sQ[256];
    sS[tid] = S;
    sQ[tid] = Q;
    __syncthreads();
#pragma unroll
    for (int s = 128; s > 0; s >>= 1) {
        if (tid < s) { sS[tid] += sS[tid + s]; sQ[tid] += sQ[tid + s]; }
        __syncthreads();
    }
    if (tid == 0) out[pair] = sQ[0] - sS[0];
}

extern "C" void kernel_launch(void* const* d_in, const int* in_sizes, int n_in,
                              void* d_out, int out_size, void* d_ws, size_t ws_size,
                              hipStream_t stream) {
    (void)in_sizes; (void)n_in; (void)out_size; (void)ws_size;

    const float* h_real = (const float*)d_in[0];
    const float* h_aug  = (const float*)d_in[1];
    const float* F_mat  = (const float*)d_in[2];
    const float* F_aug  = (const float*)d_in[3];

    float* fth_ws = (float*)d_ws;                              // 2*NB*4096 floats = 4 MB
    float* qss_ws = fth_ws + (size_t)2 * NB * (K_DIM * D_DIM); // 2*NB floats

    dim3 grid(NB, 1, 2);
    kmeans_fth_partial_kernel<<<grid, THREADS, 0, stream>>>(
        h_real, h_aug, F_mat, F_aug, fth_ws, qss_ws);

    kmeans_reduce_kernel<<<2, 256, 0, stream>>>(fth_ws, qss_ws, (float*)d_out);
}